// TurboDense_2173253452238
// MI455X (gfx1250) — compile-verified
//
#include <hip/hip_runtime.h>

typedef __attribute__((ext_vector_type(2))) float v2f;
typedef __attribute__((ext_vector_type(8))) float v8f;

#define UNITS   4096
#define IN_DIM  4096
#define LEVELS  16
#define GROUP   128
#define NGROUPS 32
#define BATCH   16

#define WAVES_PER_WG     16
#define THREADS_PER_WG   (WAVES_PER_WG * 32)          // 512
#define K_PER_WAVE       (IN_DIM / WAVES_PER_WG)      // 256
#define STEPS            (K_PER_WAVE / 4)             // 64 WMMA steps per wave
#define PF               4                            // software-pipeline depth
#define CB_STRIDE        17                           // pad to spread LDS banks

__global__ __launch_bounds__(THREADS_PER_WG)
void turbo_dense_wmma_kernel(const float* __restrict__ inputs,     // [16, 4096]
                             const float* __restrict__ codebooks,  // [4096, 16]
                             const float* __restrict__ scales,     // [4096, 32]
                             const float* __restrict__ residual,   // [4096, 4096] flat
                             const float* __restrict__ bias,       // [4096]
                             const int*   __restrict__ indices,    // [4096, 4096] flat
                             float* __restrict__ out)              // [16, 4096]
{
    __shared__ float lds_cb[16 * CB_STRIDE];              // codebook tile (16ch x 16lvl)
    __shared__ float lds_red[WAVES_PER_WG * 256];         // 16 partial 16x16 C tiles

    const int tile = blockIdx.x;          // channels [tile*16, tile*16+16)
    const int tid  = threadIdx.x;         // 0..511
    const int wave = tid >> 5;            // 0..15 (wave32)
    const int lane = tid & 31;

    // Stage codebook tile into LDS: thread t -> channel t/16, level t%16
    if (tid < 256) {
        const int ch  = tid >> 4;
        const int lvl = tid & 15;
        lds_cb[ch * CB_STRIDE + lvl] = codebooks[(tile * 16 + ch) * LEVELS + lvl];
    }
    __syncthreads();

    // WMMA 16x16x4 f32 fragment mapping (wave32):
    //   lane l: M/N = l&15 ; K pair = k0 + 2*(l>>4) , +1
    const int nloc  = lane & 15;                  // N within tile (B) == M row (A)
    const int nglob = tile * 16 + nloc;           // global output channel (B row)
    const int koff  = (lane >> 4) * 2;            // 0 or 2

    const float* cb     = &lds_cb[nloc * CB_STRIDE];
    const float* inA    = inputs   + (size_t)nloc  * IN_DIM;   // A row (M = nloc)
    const int*   idxRow = indices  + (size_t)nglob * IN_DIM;   // B channel row
    const float* resRow = residual + (size_t)nglob * IN_DIM;

    const int kbase = wave * K_PER_WAVE;
    const int base  = kbase + koff;

    // Two scale groups per wave (K_PER_WAVE = 256 = 2 * GROUP), preloaded.
    const int gbase = kbase >> 7;                  // = wave * 2
    const float sg0 = scales[nglob * NGROUPS + gbase + 0];
    const float sg1 = scales[nglob * NGROUPS + gbase + 1];

    // ---- depth-PF software pipeline over fully-unrolled step loop ----
    v2f    aP[PF];
    int2   iP[PF];
    float2 rP[PF];

    #pragma unroll
    for (int p = 0; p < PF - 1; ++p) {
        const int kk = base + p * 4;
        iP[p] = *(const int2*)  (idxRow + kk);
        rP[p] = *(const float2*)(resRow + kk);
        aP[p] = *(const v2f*)   (inA    + kk);
    }

    v8f c = {};

    #pragma unroll
    for (int step = 0; step < STEPS; ++step) {
        const int slot = step % PF;

        // prefetch loads for step + PF-1 (indices are compile-time constants)
        const int pfs = step + PF - 1;
        if (pfs < STEPS) {
            const int pslot = pfs % PF;
            const int kk    = base + pfs * 4;
            iP[pslot] = *(const int2*)  (idxRow + kk);
            rP[pslot] = *(const float2*)(resRow + kk);
            aP[pslot] = *(const v2f*)   (inA    + kk);
        }

        // dequantize two K for this lane's channel (issued PF-1 steps ago)
        const float s = (step < STEPS / 2) ? sg0 : sg1;   // resolved at compile time
        v2f b;
        b.x = cb[iP[slot].x] * s + rP[slot].x;
        b.y = cb[iP[slot].y] * s + rP[slot].y;

        // D(16x16 f32) += A(16x4 f32) x B(4x16 f32)
        c = __builtin_amdgcn_wmma_f32_16x16x4_f32(
                /*neg_a=*/false, aP[slot], /*neg_b=*/false, b,
                /*c_mod=*/(short)0, c, /*reuse_a=*/false, /*reuse_b=*/false);
    }

    // ---- cross-wave reduction of partial C tiles ----
    #pragma unroll
    for (int v = 0; v < 8; ++v)
        lds_red[wave * 256 + v * 32 + lane] = c[v];
    __syncthreads();

    if (tid < 256) {
        float acc = 0.0f;
        #pragma unroll
        for (int w = 0; w < WAVES_PER_WG; ++w)
            acc += lds_red[w * 256 + tid];

        // C layout: element (vgpr v, lane l) -> M = v + 8*(l>>4), N = l&15
        const int v = tid >> 5;
        const int l = tid & 31;
        const int M = v + 8 * (l >> 4);
        const int N = l & 15;
        const int ncol = tile * 16 + N;
        out[(size_t)M * UNITS + ncol] = acc + bias[ncol];
    }
}

extern "C" void kernel_launch(void* const* d_in, const int* in_sizes, int n_in,
                              void* d_out, int out_size, void* d_ws, size_t ws_size,
                              hipStream_t stream) {
    const float* inputs    = (const float*)d_in[0];
    const float* codebooks = (const float*)d_in[1];
    const float* scales    = (const float*)d_in[2];
    const float* residual  = (const float*)d_in[3];
    const float* bias      = (const float*)d_in[4];
    const int*   indices   = (const int*)d_in[5];
    float*       out       = (float*)d_out;

    dim3 grid(UNITS / 16);          // 256 channel tiles
    dim3 block(THREADS_PER_WG);     // 16 wave32s
    turbo_dense_wmma_kernel<<<grid, block, 0, stream>>>(
        inputs, codebooks, scales, residual, bias, indices, out);
}